// SpatialEmbedding_46342697124026
// MI455X (gfx1250) — compile-verified
//
#include <hip/hip_runtime.h>
#include <hip/hip_bf16.h>

typedef float v2f __attribute__((ext_vector_type(2)));
typedef float v8f __attribute__((ext_vector_type(8)));

#define B_    2
#define N_    384
#define K_    256
#define D_    512
#define NT    16            // WMMA tile edge
#define NJT   (N_ / NT)     // 24 tiles along j (and i)
#define KC    4             // k-chunks per (b, i-tile)
#define KPC   (K_ / KC)     // 64 wavenumbers per block
#define EPS   1e-6f
#define INV_2PI 0.15915494309189535f

__global__ __launch_bounds__(256)
void SpatialEmbedding_gfx1250_kernel(const float* __restrict__ coords,
                                     const float* __restrict__ wavenumbers,
                                     const int*   __restrict__ pad,
                                     float*       __restrict__ out)
{
    // LDS: staged batch coords/norms/pad + interleaved {r, 1/r} tile (~57 KB)
    __shared__ float  s_x[N_], s_y[N_], s_z[N_], s_n[N_];
    __shared__ int    s_pad[N_];
    __shared__ float2 s_rw[NT][N_];     // .x = r, .y = masked 1/r

    const int tid = threadIdx.x;
    int bid = blockIdx.x;
    const int kc = bid % KC;  bid /= KC;
    const int it = bid % NJT; bid /= NJT;
    const int b  = bid;                       // batch index 0..B_-1

    // ---- Stage coords of batch b into LDS ----
    for (int j = tid; j < N_; j += 256) {
        const float x = coords[(b * N_ + j) * 3 + 0];
        const float y = coords[(b * N_ + j) * 3 + 1];
        const float z = coords[(b * N_ + j) * 3 + 2];
        s_x[j] = x; s_y[j] = y; s_z[j] = z;
        s_n[j] = x * x + y * y + z * z;
        s_pad[j] = pad[b * N_ + j];
    }
    __syncthreads();

    // ---- Phase A: Gram-matrix distances via V_WMMA_F32_16X16X4_F32 ----
    const int lane = tid & 31;
    const int wave = tid >> 5;      // 0..7
    const int half = lane >> 4;     // 0 = lanes 0-15, 1 = lanes 16-31
    const int lidx = lane & 15;
    const int ibase = it * NT;

    // A tile (16x4): lanes 0-15 -> (x,y) K=0,1 ; lanes 16-31 -> (z,0) K=2,3
    const float ax = half ? s_z[ibase + lidx] : s_x[ibase + lidx];
    const float ay = half ? 0.0f              : s_y[ibase + lidx];
    v2f a; a.x = ax; a.y = ay;

    for (int jt = wave; jt < NJT; jt += 8) {   // 3 j-tiles per wave, uniform trip count
        const int jbase = jt * NT;
        const float bx = half ? s_z[jbase + lidx] : s_x[jbase + lidx];
        const float by = half ? 0.0f              : s_y[jbase + lidx];
        v2f bm; bm.x = bx; bm.y = by;

        v8f c = {};
        // D = A x B (dot products of coordinate triples, K padded to 4)
        c = __builtin_amdgcn_wmma_f32_16x16x4_f32(
                /*neg_a=*/false, a, /*neg_b=*/false, bm,
                /*c_mod=*/(short)0, c, /*reuse_a=*/false, /*reuse_b=*/false);

        const int   col    = lidx;
        const int   jg     = jbase + col;
        const float nj     = s_n[jg];
        const bool  jvalid = (s_pad[jg] == 0);
        #pragma unroll
        for (int v = 0; v < 8; ++v) {
            const int   row = v + 8 * half;      // D layout: M = v + 8*(lane>=16)
            const int   ig  = ibase + row;
            float r2 = s_n[ig] + nj - 2.0f * c[v];
            r2 = fmaxf(r2, 0.0f) + 1e-12f;
            const float r = __builtin_amdgcn_sqrtf(r2);      // raw v_sqrt_f32 (~1 ulp)
            const bool valid = jvalid && (ig != jg) && (r > EPS);
            float2 rw;
            rw.x = r;
            rw.y = valid ? __builtin_amdgcn_rcpf(r) : 0.0f;  // raw v_rcp_f32 (~1 ulp)
            s_rw[row][jg] = rw;
        }
    }
    __syncthreads();

    // ---- Phase B: transcendental reduction over j (the real cost: ~151M v_sin/v_cos) ----
    const int   klocal = tid & (KPC - 1);        // 0..63
    const int   k      = kc * KPC + klocal;
    const float wk_rev = wavenumbers[k] * INV_2PI;  // hoist 1/(2pi): hw sin computes sin(2pi*S)

    for (int il = (tid >> 6); il < NT; il += 4) {   // 4 target rows per thread
        float acc_re = 0.0f, acc_im = 0.0f;
        #pragma unroll 4
        for (int j = 0; j < N_; ++j) {               // wave-uniform LDS address -> broadcast
            const float2 rw = s_rw[il][j];           // one ds_load_b64
            const float  t  = wk_rev * rw.x;         // phase in revolutions
            acc_re = fmaf(__builtin_amdgcn_cosf(t), rw.y, acc_re);   // v_cos_f32
            acc_im = fmaf(__builtin_amdgcn_sinf(t), rw.y, acc_im);   // v_sin_f32
        }
        const int  ig   = ibase + il;
        const bool tpad = (s_pad[ig] != 0);          // zero padded target rows
        float2 res;
        res.x = tpad ? 0.0f : acc_re;
        res.y = tpad ? 0.0f : acc_im;
        *(float2*)(out + (size_t)(b * N_ + ig) * D_ + 2 * k) = res;  // global_store_b64
    }
}

extern "C" void kernel_launch(void* const* d_in, const int* in_sizes, int n_in,
                              void* d_out, int out_size, void* d_ws, size_t ws_size,
                              hipStream_t stream) {
    (void)in_sizes; (void)n_in; (void)d_ws; (void)ws_size; (void)out_size;
    const float* coords      = (const float*)d_in[0];   // [2,384,3] f32
    const float* wavenumbers = (const float*)d_in[1];   // [256] f32
    const int*   pad         = (const int*)d_in[2];     // [2,384] (True = padding)
    float*       out         = (float*)d_out;           // [2,384,512] f32

    const int grid = B_ * NJT * KC;                     // 192 blocks
    SpatialEmbedding_gfx1250_kernel<<<grid, 256, 0, stream>>>(coords, wavenumbers, pad, out);
}